// Featuration_75969381532370
// MI455X (gfx1250) — compile-verified
//
#include <hip/hip_runtime.h>

typedef __attribute__((ext_vector_type(2))) float v2f;
typedef __attribute__((ext_vector_type(8))) float v8f;

#define BB 8
#define NN 8192
#define MM 128
#define TILES_PER_BATCH (NN / 16)   // 512
#define WAVES_PER_BLOCK 8

__global__ __launch_bounds__(32 * WAVES_PER_BLOCK)
void featuration_kernel(const float* __restrict__ xyz,   // [B,3,N]
                        const float* __restrict__ V,     // [B,M,3]
                        const float* __restrict__ W1,    // [4,4]
                        const float* __restrict__ b1,    // [4]
                        const float* __restrict__ W2,    // [4,4]
                        const float* __restrict__ b2,    // [4]
                        const float* __restrict__ W3,    // [1,4]
                        const float* __restrict__ b3,    // [1]
                        float* __restrict__ out)         // [B,N,M]
{
    const int lane = threadIdx.x & 31;
    const int wave = threadIdx.x >> 5;
    const int g    = blockIdx.x * WAVES_PER_BLOCK + wave;   // 16-point tile id
    const int b    = g / TILES_PER_BATCH;
    const int n0   = (g % TILES_PER_BATCH) * 16;

    const int half = lane >> 4;    // 0: K={0,1}, C-rows 0..7 ; 1: K={2,3}, C-rows 8..15
    const int ml   = lane & 15;    // A-row index and C-column (anchor) index

    const float* xb = xyz + (size_t)b * 3 * NN;
    const float* Vb = V   + (size_t)b * MM * 3;

    // ---- uniform weights -> scalar registers ----
    float w1[16], w2[16], w3[4], bb1[4], bb2[4];
    #pragma unroll
    for (int i = 0; i < 16; ++i) { w1[i] = W1[i]; w2[i] = W2[i]; }
    #pragma unroll
    for (int i = 0; i < 4; ++i)  { bb1[i] = b1[i]; bb2[i] = b2[i]; w3[i] = W3[i]; }
    const float bb3 = b3[0];

    // ---- A operand: 16 points x K=4 (K=3 zero-padded), f32 16x4 layout ----
    v2f a;
    {
        const int na = n0 + ml;
        float k02 = xb[(half ? 2 : 0) * NN + na];   // VGPR0: K=0 / K=2
        float k1  = xb[1 * NN + na];                // VGPR1: K=1 / K=3 -> 0
        a.x = k02;
        a.y = half ? 0.0f : k1;
    }

    // ---- per-row squared norms |p|^2 for this lane's 8 C-rows ----
    float pp[8];
    #pragma unroll
    for (int r = 0; r < 8; ++r) {
        const int nr = n0 + r + 8 * half;
        float x = xb[0 * NN + nr];
        float y = xb[1 * NN + nr];
        float z = xb[2 * NN + nr];
        pp[r] = __builtin_fmaf(x, x, __builtin_fmaf(y, y, z * z));
    }

    // ---- 8 WMMAs; convert dot -> distance in-place; hoist m-only layer-1 part ----
    float dd[8][8];     // dd[t][r] = distance, anchor m = 16t+ml, point row r+8*half
    float p1h[8][4];    // layer-1 partial: b1[j] + W1[j,1..3] . V[m]

    #pragma unroll
    for (int t = 0; t < 8; ++t) {
        const int m = t * 16 + ml;
        const float vxm = Vb[m * 3 + 0];
        const float vym = Vb[m * 3 + 1];
        const float vzm = Vb[m * 3 + 2];

        v2f bv;                                   // B: 4x16 tile of V^T
        bv.x = half ? vzm : vxm;                  // VGPR0: K=0 / K=2
        bv.y = half ? 0.0f : vym;                 // VGPR1: K=1 / K=3 -> 0

        v8f acc = {};
        acc = __builtin_amdgcn_wmma_f32_16x16x4_f32(
            false, a, false, bv, (short)0, acc, false, false);

        const float vv =
            __builtin_fmaf(vxm, vxm, __builtin_fmaf(vym, vym, vzm * vzm));

        #pragma unroll
        for (int j = 0; j < 4; ++j) {
            p1h[t][j] = __builtin_fmaf(w1[j * 4 + 1], vxm,
                        __builtin_fmaf(w1[j * 4 + 2], vym,
                        __builtin_fmaf(w1[j * 4 + 3], vzm, bb1[j])));
        }

        #pragma unroll
        for (int r = 0; r < 8; ++r) {
            float d2 = __builtin_fmaf(-2.0f, acc[r], pp[r]) + vv;
            dd[t][r] = __builtin_amdgcn_sqrtf(fmaxf(d2, 0.0f));
        }
    }

    // ---- per-row min/max over 128 anchors, normalize, MLP, store ----
    float* op = out + ((size_t)(b * NN + n0 + 8 * half)) * MM + ml;  // lane base

    #pragma unroll
    for (int r = 0; r < 8; ++r) {
        float mn = dd[0][r], mx = dd[0][r];
        #pragma unroll
        for (int t = 1; t < 8; ++t) {
            mn = fminf(mn, dd[t][r]);
            mx = fmaxf(mx, dd[t][r]);
        }
        // butterfly across the 16-lane column group (masks stay inside each half)
        #pragma unroll
        for (int s = 1; s <= 8; s <<= 1) {
            mn = fminf(mn, __shfl_xor(mn, s, 32));
            mx = fmaxf(mx, __shfl_xor(mx, s, 32));
        }
        const float denom = mx - mn;
        const float inv   = denom > 0.0f ? 1.0f / denom : 0.0f;  // 0 -> f==0 (NaN mask)
        const float nmn   = -mn * inv;

        #pragma unroll
        for (int t = 0; t < 8; ++t) {
            const float f = __builtin_fmaf(dd[t][r], inv, nmn);

            float h[4];
            #pragma unroll
            for (int j = 0; j < 4; ++j) {
                float u = __builtin_fmaf(w1[j * 4 + 0], f, p1h[t][j]);
                h[j] = fmaxf(u, 0.2f * u);                 // leaky(x)=max(x,0.2x)
            }
            float h2[4];
            #pragma unroll
            for (int j = 0; j < 4; ++j) {
                float u = __builtin_fmaf(w2[j * 4 + 0], h[0],
                          __builtin_fmaf(w2[j * 4 + 1], h[1],
                          __builtin_fmaf(w2[j * 4 + 2], h[2],
                          __builtin_fmaf(w2[j * 4 + 3], h[3], bb2[j]))));
                h2[j] = fmaxf(u, 0.2f * u);
            }
            const float o = __builtin_fmaf(w3[0], h2[0],
                            __builtin_fmaf(w3[1], h2[1],
                            __builtin_fmaf(w3[2], h2[2],
                            __builtin_fmaf(w3[3], h2[3], bb3))));

            op[r * MM + t * 16] = o;   // constant byte offset r*512 + t*64 -> IOFFSET
        }
    }
}

extern "C" void kernel_launch(void* const* d_in, const int* in_sizes, int n_in,
                              void* d_out, int out_size, void* d_ws, size_t ws_size,
                              hipStream_t stream) {
    (void)in_sizes; (void)n_in; (void)out_size; (void)d_ws; (void)ws_size;

    const float* xyz = (const float*)d_in[0];
    const float* V   = (const float*)d_in[1];
    const float* W1  = (const float*)d_in[2];
    const float* b1  = (const float*)d_in[3];
    const float* W2  = (const float*)d_in[4];
    const float* b2  = (const float*)d_in[5];
    const float* W3  = (const float*)d_in[6];
    const float* b3  = (const float*)d_in[7];
    float* out = (float*)d_out;

    const int total_tiles = BB * TILES_PER_BATCH;           // 4096
    const int blocks = total_tiles / WAVES_PER_BLOCK;       // 512
    featuration_kernel<<<blocks, 32 * WAVES_PER_BLOCK, 0, stream>>>(
        xyz, V, W1, b1, W2, b2, W3, b3, out);
}